// Model_12558484373506
// MI455X (gfx1250) — compile-verified
//
#include <hip/hip_runtime.h>
#include <stdint.h>

typedef __attribute__((ext_vector_type(2))) float v2f;
typedef __attribute__((ext_vector_type(8))) float v8f;

#define HID     64
#define CINCH   116
#define TSTEPS  12
#define HGT     6
#define WID     11
#define HWC     66          // 6*11
#define BATCH   64
#define NGRU    (BATCH*HWC) // 4224
#define KGRU    ((CINCH+HID)*9) // 1620  (= 135 * 12)
#define KXPART  (CINCH*9)   // 1044
#define DMODEL  128
#define NVAR    4228        // 4224 hidden-variates + 4 mark rows
#define L0      4228
#define L1      2114
#define L2      1057
#define SLOT    (HID*NGRU)  // 270336 floats, one hidden-state slab

#define KSLAB   12          // K rows staged per LDS slab (divides 1620 exactly)
#define SBS     68          // padded LDS row stride (floats) to dodge bank conflicts

static __device__ __forceinline__ float sigm(float x) { return 1.0f / (1.0f + expf(-x)); }

// ---------------------------------------------------------------------------
// fp32 WMMA GEMM with async Global->LDS double-buffered B staging.
//   C[M][N] = A[M][K] * B[K][N] + bias[M]
// Block = 128 threads (4 waves). Block tile = 64(M) x 64(N); each wave owns a
// 16(M) x 64(N) strip (4 accumulators). The 12x64 B k-slab is shared by all 4
// waves: staged once per block via GLOBAL_LOAD_ASYNC_TO_LDS_B128 (ASYNCcnt),
// double-buffered so the DMA of slab k+1 overlaps WMMA on slab k.
// Requires: M%64==0, N%64==0, K%12==0 (true for the ConvGRU GEMMs).
// ---------------------------------------------------------------------------
__global__ void gemm_wmma_f32_lds(const float* __restrict__ A,
                                  const float* __restrict__ B,
                                  const float* __restrict__ bias,
                                  float* __restrict__ C,
                                  int M, int N, int K)
{
    __shared__ float sb[2][KSLAB * SBS];

    const int tid  = threadIdx.x;        // 0..127
    const int wave = tid >> 5;           // 0..3
    const int lane = tid & 31;
    const int ngroups = N >> 6;
    const int mg = blockIdx.x / ngroups;
    const int ng = blockIdx.x - mg * ngroups;
    const int m0 = mg * 64 + wave * 16;
    const int n0 = ng * 64;
    const int half = lane >> 4;          // K sub-slice within a 4-step
    const int l16  = lane & 15;

    const uint32_t ldsb[2] = { (uint32_t)(uintptr_t)&sb[0][0],
                               (uint32_t)(uintptr_t)&sb[1][0] };

    // Stage one 12x64 slab (192 b128 chunks). Waves 0-1: 2 chunks/thread,
    // waves 2-3: 1 chunk/thread (wave-uniform split, EXEC stays full).
    auto stage = [&](int kbase, int bufi) {
        {
            const int c = tid;                      // chunk 0..127
            const int r = c >> 4, col = (c & 15) << 2;
            const float* g = B + (size_t)(kbase + r) * N + n0 + col;
            const uint32_t l = ldsb[bufi] + (uint32_t)((r * SBS + col) * 4);
            asm volatile("global_load_async_to_lds_b128 %0, %1, off"
                         :: "v"(l), "v"(g) : "memory");
        }
        if (wave < 2) {
            const int c = tid + 128;                // chunk 128..191
            const int r = c >> 4, col = (c & 15) << 2;
            const float* g = B + (size_t)(kbase + r) * N + n0 + col;
            const uint32_t l = ldsb[bufi] + (uint32_t)((r * SBS + col) * 4);
            asm volatile("global_load_async_to_lds_b128 %0, %1, off"
                         :: "v"(l), "v"(g) : "memory");
        }
    };

    v8f z = {};
    v8f acc0 = z, acc1 = z, acc2 = z, acc3 = z;

    const float* Arow = A + (size_t)(m0 + l16) * K + 2 * half;   // A[m][k + 2*half]

    stage(0, 0);
    int buf = 0;
    for (int k = 0; k < K; k += KSLAB) {
        const bool more = (k + KSLAB) < K;
        if (more) {
            stage(k + KSLAB, buf ^ 1);              // prefetch next slab
            // wait only for the older slab; newest stage stays in flight
            if (wave < 2) asm volatile("s_wait_asynccnt 0x2" ::: "memory");
            else          asm volatile("s_wait_asynccnt 0x1" ::: "memory");
        } else {
            asm volatile("s_wait_asynccnt 0x0" ::: "memory");
        }
        __syncthreads();                            // slab visible block-wide

        const float* sbuf = &sb[buf][0];
        #pragma unroll
        for (int kk = 0; kk < KSLAB; kk += 4) {
            v2f a;
            a.x = Arow[k + kk];
            a.y = Arow[k + kk + 1];
            const float* sp = sbuf + (kk + 2 * half) * SBS + l16;
            v2f b0, b1, b2, b3;
            b0.x = sp[0];       b0.y = sp[SBS];
            b1.x = sp[16];      b1.y = sp[SBS + 16];
            b2.x = sp[32];      b2.y = sp[SBS + 32];
            b3.x = sp[48];      b3.y = sp[SBS + 48];
            acc0 = __builtin_amdgcn_wmma_f32_16x16x4_f32(false, a, false, b0, (short)0, acc0, false, false);
            acc1 = __builtin_amdgcn_wmma_f32_16x16x4_f32(false, a, false, b1, (short)0, acc1, false, false);
            acc2 = __builtin_amdgcn_wmma_f32_16x16x4_f32(false, a, false, b2, (short)0, acc2, false, false);
            acc3 = __builtin_amdgcn_wmma_f32_16x16x4_f32(false, a, false, b3, (short)0, acc3, false, false);
        }
        __syncthreads();                            // done reading before restage
        buf ^= 1;
    }

    #pragma unroll
    for (int r = 0; r < 8; ++r) {
        const int m = m0 + r + 8 * half;            // C/D layout: row = r + 8*(lane>=16)
        const float bv = bias ? bias[m] : 0.0f;
        float* Cp = C + (size_t)m * N + n0 + l16;
        Cp[0]  = acc0[r] + bv;
        Cp[16] = acc1[r] + bv;
        Cp[32] = acc2[r] + bv;
        Cp[48] = acc3[r] + bv;
    }
}

// ---------------------------------------------------------------------------
// Implicit-GEMM 1D conv via WMMA fp32:
//   Y[b][o][v] = sum_{c,kk} W[o][c*KS+kk] * X[b][c][v+kk-PAD] + bias[o]
// One wave: one batch b, 16 output channels, 64 output positions.
// ---------------------------------------------------------------------------
template <int KS, int PAD>
__global__ void conv1d_wmma(const float* __restrict__ X,
                            const float* __restrict__ W,
                            const float* __restrict__ bias,
                            float* __restrict__ Y,
                            int Bn, int C, int L, int O)
{
    const int wave = (blockIdx.x * blockDim.x + threadIdx.x) >> 5;
    const int lane = threadIdx.x & 31;
    const int mtiles  = O >> 4;
    const int ngroups = (L + 63) >> 6;
    const int wpb = mtiles * ngroups;
    if (wave >= Bn * wpb) return;                   // uniform per-wave exit
    const int b   = wave / wpb;
    const int rem = wave - b * wpb;
    const int mt  = rem / ngroups;
    const int ng  = rem - mt * ngroups;
    const int m0 = mt << 4;
    const int n0 = ng << 6;
    const int half = lane >> 4;
    const int l16  = lane & 15;
    const int K = C * KS;

    v8f z = {};
    v8f acc0 = z, acc1 = z, acc2 = z, acc3 = z;

    const float* Arow = W + (size_t)(m0 + l16) * K + 2 * half;
    const float* Xb   = X + (size_t)b * C * L;
    const int v0 = n0 + l16;

    for (int k = 0; k < K; k += 4) {
        v2f a;
        a.x = Arow[k];
        a.y = Arow[k + 1];
        const int k1 = k + 2 * half;
        const int c1 = k1 / KS, q1 = k1 - c1 * KS;
        const int k2 = k1 + 1;
        const int c2 = k2 / KS, q2 = k2 - c2 * KS;
        const float* Xc1 = Xb + (size_t)c1 * L;
        const float* Xc2 = Xb + (size_t)c2 * L;
        const int d1 = q1 - PAD, d2 = q2 - PAD;
        #pragma unroll
        for (int i = 0; i < 4; ++i) {
            const int v = v0 + i * 16;
            const int v1 = v + d1;
            const int v2 = v + d2;
            v2f bb;
            bb.x = (v1 >= 0 && v1 < L) ? Xc1[v1] : 0.0f;
            bb.y = (v2 >= 0 && v2 < L) ? Xc2[v2] : 0.0f;
            if (i == 0) acc0 = __builtin_amdgcn_wmma_f32_16x16x4_f32(false, a, false, bb, (short)0, acc0, false, false);
            if (i == 1) acc1 = __builtin_amdgcn_wmma_f32_16x16x4_f32(false, a, false, bb, (short)0, acc1, false, false);
            if (i == 2) acc2 = __builtin_amdgcn_wmma_f32_16x16x4_f32(false, a, false, bb, (short)0, acc2, false, false);
            if (i == 3) acc3 = __builtin_amdgcn_wmma_f32_16x16x4_f32(false, a, false, bb, (short)0, acc3, false, false);
        }
    }

    float* Yb = Y + (size_t)b * O * L;
    #pragma unroll
    for (int r = 0; r < 8; ++r) {
        const int m = m0 + r + 8 * half;
        const float bv = bias[m];
        float* Yp = Yb + (size_t)m * L;
        if (v0      < L) Yp[v0]      = acc0[r] + bv;
        if (v0 + 16 < L) Yp[v0 + 16] = acc1[r] + bv;
        if (v0 + 32 < L) Yp[v0 + 32] = acc2[r] + bv;
        if (v0 + 48 < L) Yp[v0 + 48] = acc3[r] + bv;
    }
}

// ---------------------------------------------------------------------------
// im2col for the ConvGRU 3x3 SAME conv. col[k][n], k = ci*9+kh*3+kw,
// n = b*66 + h*11 + w. ci<116 reads x_enc step t; ci>=116 reads hsrc.
// kstart=0 builds everything; kstart=KXPART rebuilds only the hidden rows.
// ---------------------------------------------------------------------------
__global__ void im2col_gru(const float* __restrict__ xe, int t,
                           const float* __restrict__ hsrc,
                           float* __restrict__ col, int kstart)
{
    const int idx = blockIdx.x * blockDim.x + threadIdx.x;
    const int total = (KGRU - kstart) * NGRU;
    if (idx >= total) return;
    const int k = kstart + idx / NGRU;
    const int n = idx - (k - kstart) * NGRU;
    const int ci = k / 9;
    const int j  = k - ci * 9;
    const int kh = j / 3, kw = j - kh * 3;
    const int b  = n / HWC;
    const int hw = n - b * HWC;
    const int h  = hw / WID, w = hw - h * WID;
    const int hh = h + kh - 1, ww = w + kw - 1;
    float v = 0.0f;
    if (hh >= 0 && hh < HGT && ww >= 0 && ww < WID) {
        const int hw2 = hh * WID + ww;
        if (ci < CINCH)
            v = xe[((size_t)b * TSTEPS + t) * (CINCH * HWC) + ci * HWC + hw2];
        else
            v = hsrc[(size_t)(ci - CINCH) * NGRU + b * HWC + hw2];
    }
    col[(size_t)k * NGRU + n] = v;
}

__global__ void zero_f32(float* p, int n)
{
    int i = blockIdx.x * blockDim.x + threadIdx.x;
    if (i < n) p[i] = 0.0f;
}

__global__ void gru_rh(const float* __restrict__ gates,
                       const float* __restrict__ hp,
                       float* __restrict__ rh)
{
    int i = blockIdx.x * blockDim.x + threadIdx.x;
    if (i >= SLOT) return;
    rh[i] = sigm(gates[i]) * hp[i];
}

__global__ void gru_hnew(const float* __restrict__ gates,
                         const float* __restrict__ cand,
                         const float* __restrict__ hp,
                         float* __restrict__ hn)
{
    int i = blockIdx.x * blockDim.x + threadIdx.x;
    if (i >= SLOT) return;
    const float u = sigm(gates[SLOT + i]);
    const float c = tanhf(cand[i]);
    hn[i] = (1.0f - u) * hp[i] + u * c;
}

// enc[b][d][v] = sum_t tokens[b][v][t]*Wemb[t][d] + bemb[d]
__global__ void embed_enc(const float* __restrict__ hs,
                          const float* __restrict__ xmark,
                          const float* __restrict__ Wemb,
                          const float* __restrict__ bemb,
                          float* __restrict__ enc)
{
    const int idx = blockIdx.x * blockDim.x + threadIdx.x;
    const int total = BATCH * DMODEL * NVAR;
    if (idx >= total) return;
    const int b = idx / (DMODEL * NVAR);
    const int r = idx - b * (DMODEL * NVAR);
    const int d = r / NVAR;
    const int v = r - d * NVAR;
    float s = bemb[d];
    if (v < NGRU) {
        const int ci = v / HWC;
        const int hw = v - ci * HWC;
        const float* hp = hs + (size_t)ci * NGRU + b * HWC + hw;
        #pragma unroll
        for (int t = 0; t < TSTEPS; ++t)
            s += hp[(size_t)t * SLOT] * Wemb[t * DMODEL + d];
    } else {
        const int j = v - NGRU;
        #pragma unroll
        for (int t = 0; t < TSTEPS; ++t)
            s += xmark[(b * TSTEPS + t) * 4 + j] * Wemb[t * DMODEL + d];
    }
    enc[((size_t)b * DMODEL + d) * NVAR + v] = s;
}

// stride-2 pool over last dim. mode: 1=max, 0=avg(*0.5). optional relu.
__global__ void pool1d(const float* __restrict__ X, float* __restrict__ Y,
                       int BC, int Lout, int mode, int relu)
{
    const int idx = blockIdx.x * blockDim.x + threadIdx.x;
    if (idx >= BC * Lout) return;
    const int v  = idx % Lout;
    const int bc = idx / Lout;
    const float* p = X + (size_t)bc * (Lout * 2) + v * 2;
    const float a = p[0], b = p[1];
    float r = mode ? fmaxf(a, b) : 0.5f * (a + b);
    if (relu) r = fmaxf(r, 0.0f);
    Y[idx] = r;
}

// out[b][i][v] = p2[b][20+i][v] + sum_c p1[b][c][v]*Wr[c][20+i] + br[20+i]
__global__ void final_out(const float* __restrict__ p1,
                          const float* __restrict__ p2,
                          const float* __restrict__ Wr,
                          const float* __restrict__ br,
                          float* __restrict__ out)
{
    const int idx = blockIdx.x * blockDim.x + threadIdx.x;
    const int total = BATCH * 12 * L2;
    if (idx >= total) return;
    const int v = idx % L2;
    const int r = idx / L2;
    const int i = r % 12;
    const int b = r / 12;
    const int o = 20 + i;
    const float* p1b = p1 + (size_t)b * 32 * L2;
    float s = br[o];
    #pragma unroll
    for (int c = 0; c < 32; ++c)
        s += p1b[(size_t)c * L2 + v] * Wr[c * 32 + o];
    out[idx] = p2[((size_t)b * 32 + o) * L2 + v] + s;
}

// ---------------------------------------------------------------------------
static inline void run_gemm(const float* A, const float* B, const float* bias,
                            float* C, int M, int N, int K, hipStream_t s)
{
    const int blocks = (M >> 6) * (N >> 6);         // 64x64 block tiles
    gemm_wmma_f32_lds<<<blocks, 128, 0, s>>>(A, B, bias, C, M, N, K);
}

extern "C" void kernel_launch(void* const* d_in, const int* in_sizes, int n_in,
                              void* d_out, int out_size, void* d_ws, size_t ws_size,
                              hipStream_t stream)
{
    const float* x_enc  = (const float*)d_in[0];
    const float* x_mark = (const float*)d_in[1];
    const float* Wg   = (const float*)d_in[4];
    const float* bg   = (const float*)d_in[5];
    const float* Wc   = (const float*)d_in[6];
    const float* bc   = (const float*)d_in[7];
    const float* Wemb = (const float*)d_in[8];
    const float* bemb = (const float*)d_in[9];
    const float* W1   = (const float*)d_in[10];
    const float* b1   = (const float*)d_in[11];
    const float* W2   = (const float*)d_in[12];
    const float* b2   = (const float*)d_in[13];
    const float* Wp1  = (const float*)d_in[14];
    const float* bp1  = (const float*)d_in[15];
    const float* Wp2  = (const float*)d_in[16];
    const float* bp2  = (const float*)d_in[17];
    const float* Wr   = (const float*)d_in[18];
    const float* br   = (const float*)d_in[19];
    float* out = (float*)d_out;
    float* ws  = (float*)d_ws;

    // workspace layout (floats)
    const size_t o_hs    = 0;                              // 12 * SLOT
    const size_t o_h0    = o_hs + (size_t)TSTEPS * SLOT;   // SLOT
    const size_t o_col   = o_h0 + SLOT;                    // KGRU * NGRU
    const size_t o_gates = o_col + (size_t)KGRU * NGRU;    // 128 * NGRU
    const size_t o_rh    = o_gates + (size_t)2 * SLOT;     // SLOT
    const size_t o_cand  = o_rh + SLOT;                    // SLOT
    const size_t o_enc   = o_col;                          // reuse GRU scratch after loop
    const size_t o_pa    = o_enc + (size_t)BATCH * DMODEL * NVAR;  // B*32*L0
    const size_t o_pb    = o_pa + (size_t)BATCH * 32 * L0;         // B*32*L1
    const size_t o_pc    = o_pb + (size_t)BATCH * 32 * L1;         // B*32*L1
    const size_t o_p1    = o_pc + (size_t)BATCH * 32 * L1;         // B*32*L2
    const size_t o_p2    = o_p1 + (size_t)BATCH * 32 * L2;         // B*32*L2

    float* hs    = ws + o_hs;
    float* h0    = ws + o_h0;
    float* col   = ws + o_col;
    float* gates = ws + o_gates;
    float* rh    = ws + o_rh;
    float* cand  = ws + o_cand;
    float* enc   = ws + o_enc;
    float* pa    = ws + o_pa;
    float* pb    = ws + o_pb;
    float* pc    = ws + o_pc;
    float* p1b   = ws + o_p1;
    float* p2b   = ws + o_p2;

    // h0 = 0
    zero_f32<<<(SLOT + 255) / 256, 256, 0, stream>>>(h0, SLOT);

    // ---- ConvGRU scan over T=12 steps ----
    for (int t = 0; t < TSTEPS; ++t) {
        const float* hprev = (t == 0) ? h0 : hs + (size_t)(t - 1) * SLOT;
        float* hnext = hs + (size_t)t * SLOT;

        { // full im2col of [xt, h_prev]
            const int tot = KGRU * NGRU;
            im2col_gru<<<(tot + 255) / 256, 256, 0, stream>>>(x_enc, t, hprev, col, 0);
        }
        run_gemm(Wg, col, bg, gates, 2 * HID, NGRU, KGRU, stream); // gates = Wg*col + bg
        gru_rh<<<(SLOT + 255) / 256, 256, 0, stream>>>(gates, hprev, rh);
        { // rebuild only the hidden rows of im2col with r*h
            const int tot = (KGRU - KXPART) * NGRU;
            im2col_gru<<<(tot + 255) / 256, 256, 0, stream>>>(x_enc, t, rh, col, KXPART);
        }
        run_gemm(Wc, col, bc, cand, HID, NGRU, KGRU, stream);      // cand = Wc*col + bc
        gru_hnew<<<(SLOT + 255) / 256, 256, 0, stream>>>(gates, cand, hprev, hnext);
    }

    // ---- inverted embedding -> enc[B][128][4228] ----
    {
        const int tot = BATCH * DMODEL * NVAR;
        embed_enc<<<(tot + 255) / 256, 256, 0, stream>>>(hs, x_mark, Wemb, bemb, enc);
    }

    // ---- path 1: conv7 -> maxpool -> conv7 -> maxpool -> relu ----
    {
        const int waves = BATCH * (32 >> 4) * ((L0 + 63) >> 6);
        conv1d_wmma<7, 3><<<(waves * 32 + 127) / 128, 128, 0, stream>>>(enc, W1, b1, pa, BATCH, DMODEL, L0, 32);
        pool1d<<<(BATCH * 32 * L1 + 255) / 256, 256, 0, stream>>>(pa, pb, BATCH * 32, L1, 1, 0);
        const int waves2 = BATCH * (32 >> 4) * ((L1 + 63) >> 6);
        conv1d_wmma<7, 3><<<(waves2 * 32 + 127) / 128, 128, 0, stream>>>(pb, W2, b2, pc, BATCH, 32, L1, 32);
        pool1d<<<(BATCH * 32 * L2 + 255) / 256, 256, 0, stream>>>(pc, p1b, BATCH * 32, L2, 1, 1);
    }

    // ---- path 2: conv11 -> avgpool -> conv11 -> avgpool -> relu ----
    {
        const int waves = BATCH * (32 >> 4) * ((L0 + 63) >> 6);
        conv1d_wmma<11, 5><<<(waves * 32 + 127) / 128, 128, 0, stream>>>(enc, Wp1, bp1, pa, BATCH, DMODEL, L0, 32);
        pool1d<<<(BATCH * 32 * L1 + 255) / 256, 256, 0, stream>>>(pa, pb, BATCH * 32, L1, 0, 0);
        const int waves2 = BATCH * (32 >> 4) * ((L1 + 63) >> 6);
        conv1d_wmma<11, 5><<<(waves2 * 32 + 127) / 128, 128, 0, stream>>>(pb, Wp2, bp2, pc, BATCH, 32, L1, 32);
        pool1d<<<(BATCH * 32 * L2 + 255) / 256, 256, 0, stream>>>(pc, p2b, BATCH * 32, L2, 0, 1);
    }

    // ---- residual (transpose(p1) @ Wr + br), add p2, slice last 12 channels ----
    {
        const int tot = BATCH * 12 * L2;
        final_out<<<(tot + 255) / 256, 256, 0, stream>>>(p1b, p2b, Wr, br, out);
    }
}